// QuantSSM_79182017069511
// MI455X (gfx1250) — compile-verified
//
#include <hip/hip_runtime.h>
#include <hip/hip_bf16.h>
#include <math.h>

// ---- problem constants (from reference) ----
#define BB      2
#define LL      2048
#define MM      1024        // d_inner
#define DS      16          // d_state
#define RR      64          // dt_rank
#define NPROJ   (RR + 2*DS) // 96
#define BLT     (BB*LL)     // 4096 rows

typedef float  v2f __attribute__((ext_vector_type(2)));
typedef float  v8f __attribute__((ext_vector_type(8)));

// ---------------------------------------------------------------------------
// Kernel 1: xp = x @ x_proj_w^T   (BLT x MM) * (NPROJ x MM)^T -> (BLT x NPROJ)
// One wave (32 threads) per 16x16 output tile. fp32 WMMA, K-step = 4.
// Grid: (BLT/16) * (NPROJ/16) = 256 * 6 = 1536 waves.
// ---------------------------------------------------------------------------
__global__ __launch_bounds__(32)
void proj_xp_kernel(const float* __restrict__ x,   // (BLT, MM)
                    const float* __restrict__ w,   // (NPROJ, MM)
                    float* __restrict__ xp)        // (BLT, NPROJ)
{
    const int lane = threadIdx.x;        // 0..31 (wave32)
    const int tile = blockIdx.x;
    const int tm   = tile / (NPROJ/16);  // row-tile  0..255
    const int tn   = tile % (NPROJ/16);  // col-tile  0..5
    const int half = lane >> 4;          // 0 | 1
    const int l15  = lane & 15;

    const float* __restrict__ arow = x + (size_t)(tm*16 + l15) * MM;
    const float* __restrict__ brow = w + (size_t)(tn*16 + l15) * MM;

    v8f c = {};
    #pragma unroll 4
    for (int k0 = 0; k0 < MM; k0 += 4) {
        const int ka = k0 + half*2;      // 8B aligned -> b64 loads
        v2f a; a.x = arow[ka]; a.y = arow[ka + 1];
        v2f b; b.x = brow[ka]; b.y = brow[ka + 1];
        c = __builtin_amdgcn_wmma_f32_16x16x4_f32(
                false, a, false, b, (short)0, c, false, false);
    }

    float* __restrict__ o = xp + (size_t)(tm*16) * NPROJ + tn*16 + l15;
    #pragma unroll
    for (int r = 0; r < 8; ++r) {
        const int row = r + half*8;
        o[(size_t)row * NPROJ] = c[r];
    }
}

// ---------------------------------------------------------------------------
// Kernel 2: dt = softplus( xp[:, :RR] @ dt_proj_w^T + dt_proj_b )
// (BLT x RR) * (MM x RR)^T -> (BLT x MM).  K = 64 -> 16 WMMA steps per tile.
// Grid: (BLT/16) * (MM/16) = 256 * 64 = 16384 waves.
// ---------------------------------------------------------------------------
__global__ __launch_bounds__(32)
void dt_proj_kernel(const float* __restrict__ xp,   // (BLT, NPROJ), cols [0,64) = dt_in
                    const float* __restrict__ w,    // (MM, RR)
                    const float* __restrict__ bias, // (MM)
                    float* __restrict__ dt)         // (BLT, MM)
{
    const int lane = threadIdx.x;
    const int tile = blockIdx.x;
    const int tm   = tile / (MM/16);     // 0..255
    const int tn   = tile % (MM/16);     // 0..63
    const int half = lane >> 4;
    const int l15  = lane & 15;

    const float* __restrict__ arow = xp + (size_t)(tm*16 + l15) * NPROJ;
    const float* __restrict__ brow = w  + (size_t)(tn*16 + l15) * RR;

    v8f c = {};
    #pragma unroll
    for (int k0 = 0; k0 < RR; k0 += 4) {
        const int ka = k0 + half*2;
        v2f a; a.x = arow[ka]; a.y = arow[ka + 1];
        v2f b; b.x = brow[ka]; b.y = brow[ka + 1];
        c = __builtin_amdgcn_wmma_f32_16x16x4_f32(
                false, a, false, b, (short)0, c, false, false);
    }

    const int col = tn*16 + l15;
    const float bb = bias[col];
    #pragma unroll
    for (int r = 0; r < 8; ++r) {
        const int row = tm*16 + r + half*8;
        float v = c[r] + bb;
        // numerically stable softplus
        float sp = (v > 20.0f) ? v : log1pf(__expf(v));
        dt[(size_t)row * MM + col] = sp;
    }
}

// ---------------------------------------------------------------------------
// Kernel 3: selective scan. One thread per (b, m) channel; h[16] in registers
// (16-way ILP hides v_exp_f32 + FMA latency along the serial L chain).
// B_t / C_t are block-uniform float4 loads from the xp row (16B aligned:
// row base = r*96*4 = r*384 bytes, +64*4 = +256 bytes).
// ---------------------------------------------------------------------------
__global__ __launch_bounds__(256)
void scan_kernel(const float* __restrict__ x,       // (BLT, MM)
                 const float* __restrict__ z,       // (BLT, MM)
                 const float* __restrict__ xp,      // (BLT, NPROJ)
                 const float* __restrict__ dtb,     // (BLT, MM)
                 const float* __restrict__ A_log,   // (DS, MM)
                 const float* __restrict__ Dp_arr,  // (MM)
                 float* __restrict__ out)           // (BLT, MM)
{
    const int gid = blockIdx.x * blockDim.x + threadIdx.x;  // 0..BB*MM-1
    const int b   = gid >> 10;       // / MM
    const int m   = gid & (MM - 1);

    float Am[DS], h[DS];
    #pragma unroll
    for (int d = 0; d < DS; ++d) {
        Am[d] = -__expf(A_log[(size_t)d * MM + m]);  // A = -exp(A_log).T
        h[d]  = 0.0f;
    }
    const float Dp = Dp_arr[m];
    const size_t rowB = (size_t)b * LL;

    for (int t = 0; t < LL; ++t) {
        const size_t r = rowB + t;
        const float dtv = dtb[r * MM + m];
        const float xv  = x  [r * MM + m];
        const float zv  = z  [r * MM + m];

        const float4* __restrict__ bc =
            (const float4*)(xp + r * NPROJ + RR);   // [0..3]=B_t, [4..7]=C_t
        float Bv[DS], Cv[DS];
        #pragma unroll
        for (int i = 0; i < 4; ++i) {
            float4 vb = bc[i];
            float4 vc = bc[4 + i];
            Bv[4*i+0] = vb.x; Bv[4*i+1] = vb.y; Bv[4*i+2] = vb.z; Bv[4*i+3] = vb.w;
            Cv[4*i+0] = vc.x; Cv[4*i+1] = vc.y; Cv[4*i+2] = vc.z; Cv[4*i+3] = vc.w;
        }

        // prefetch next xp rows into cache
        if (t + 8 < LL)
            __builtin_prefetch(xp + (r + 8) * NPROJ, 0, 1);

        const float dx = dtv * xv;
        float y = 0.0f;
        #pragma unroll
        for (int d = 0; d < DS; ++d) {
            h[d] = __expf(dtv * Am[d]) * h[d] + dx * Bv[d];
            y    = fmaf(h[d], Cv[d], y);
        }
        y = fmaf(Dp, xv, y);

        const float sig = 1.0f / (1.0f + __expf(-zv));  // silu(z) = z*sigmoid(z)
        out[r * MM + m] = y * (zv * sig);
    }
}

// ---------------------------------------------------------------------------
extern "C" void kernel_launch(void* const* d_in, const int* in_sizes, int n_in,
                              void* d_out, int out_size, void* d_ws, size_t ws_size,
                              hipStream_t stream) {
    const float* x         = (const float*)d_in[0];  // (B,L,M)
    const float* z         = (const float*)d_in[1];  // (B,L,M)
    const float* x_proj_w  = (const float*)d_in[2];  // (NPROJ, M)
    const float* dt_proj_w = (const float*)d_in[3];  // (M, R)
    const float* dt_proj_b = (const float*)d_in[4];  // (M,)
    const float* A_log     = (const float*)d_in[5];  // (DS, M)
    const float* D_param   = (const float*)d_in[6];  // (M,)
    float* out = (float*)d_out;

    float* xp_ws = (float*)d_ws;                         // (BLT, NPROJ) : 1.5 MB
    float* dt_ws = xp_ws + (size_t)BLT * NPROJ;          // (BLT, MM)    : 16.8 MB

    // Phase 1: input projection (fp32 WMMA)
    proj_xp_kernel<<<(BLT/16) * (NPROJ/16), 32, 0, stream>>>(x, x_proj_w, xp_ws);

    // Phase 2: dt projection + softplus (fp32 WMMA)
    dt_proj_kernel<<<(BLT/16) * (MM/16), 32, 0, stream>>>(xp_ws, dt_proj_w,
                                                          dt_proj_b, dt_ws);

    // Phase 3: sequential selective scan
    scan_kernel<<<(BB*MM)/256, 256, 0, stream>>>(x, z, xp_ws, dt_ws,
                                                 A_log, D_param, out);
}